// DotPred_13013750907177
// MI455X (gfx1250) — compile-verified
//
#include <hip/hip_runtime.h>

// score[t,e] = sum_d (x[src]-x[dst]) / sqrt(D)
//            = (rowsum(x)[src] - rowsum(x)[dst]) / sqrt(D)
//
// Pass 1 (rowsum): streams the 100000x128 f32 table once (51.2 MB), one wave
//   per 16-row tile. Per-lane partial sums are arranged as the A-operand of
//   V_WMMA_F32_16X16X4_F32; a single WMMA vs an all-ones B reduces across the
//   two lane halves, leaving rowsum(m) replicated in every column of D.
// Pass 2 (score): S (400 KB) is L2/WGP$-resident; gathers are cheap. HBM
//   traffic = int32 indices (12 MB) + 6 MB output.

typedef __attribute__((ext_vector_type(2))) float v2f;
typedef __attribute__((ext_vector_type(8))) float v8f;
typedef __attribute__((ext_vector_type(4))) float f32x4;
typedef __attribute__((ext_vector_type(4))) int   i32x4;

#define IN_DIM 128
#define WAVES_PER_BLOCK 8

__global__ __launch_bounds__(256) void rowsum_wmma_kernel(
    const float* __restrict__ x, float* __restrict__ S, int nNodes) {
  const int lane = threadIdx.x & 31;
  const int wave = threadIdx.x >> 5;
  const int tile = blockIdx.x * WAVES_PER_BLOCK + wave;
  const int nTiles = (nNodes + 15) >> 4;
  if (tile >= nTiles) return;  // wave-uniform: EXEC stays all-1s for WMMA

  const int r    = lane & 15;   // row within tile owned by this lane
  const int half = lane >> 4;   // 0: cols ==0..3 mod 8 ; 1: cols ==4..7 mod 8
  int row = tile * 16 + r;
  if (row >= nNodes) row = nNodes - 1;  // clamp loads; stores guarded below

  const float* p = x + (size_t)row * IN_DIM + half * 4;

  // 16 x b128 loads per lane; lanes L and L+16 jointly cover row L's 128 cols.
  float partial = 0.f;
#pragma unroll
  for (int c = 0; c < IN_DIM / 8; ++c) {
    f32x4 v = *(const f32x4*)(p + c * 8);
    partial += (v.x + v.y) + (v.z + v.w);
  }

  // A-matrix 16x4 f32 layout: lane L -> {A[L][0], A[L][1]}, lane L+16 ->
  // {A[L][2], A[L][3]}. With A = {partial, 0} and B = ones:
  //   D[m,n] = partial(m, half0) + partial(m, half1) = rowsum(m)  (all n).
  v2f a; a.x = partial; a.y = 0.0f;
  v2f b; b.x = 1.0f;    b.y = 1.0f;
  v8f acc = {};
  acc = __builtin_amdgcn_wmma_f32_16x16x4_f32(
      /*neg_a=*/false, a, /*neg_b=*/false, b,
      /*c_mod=*/(short)0, acc, /*reuse_a=*/false, /*reuse_b=*/false);

  // D layout: lanes 0-15 hold rows 0..7 in VGPRs 0..7; lanes 16-31 hold rows
  // 8..15. Columns are identical, so lane 0 and lane 16 each store 8 rows.
  const int baseRow = tile * 16;
  if (baseRow + 16 <= nNodes) {
    if (lane == 0) {
      f32x4 lo = {acc[0], acc[1], acc[2], acc[3]};
      f32x4 hi = {acc[4], acc[5], acc[6], acc[7]};
      *(f32x4*)(S + baseRow)     = lo;
      *(f32x4*)(S + baseRow + 4) = hi;
    }
    if (lane == 16) {
      f32x4 lo = {acc[0], acc[1], acc[2], acc[3]};
      f32x4 hi = {acc[4], acc[5], acc[6], acc[7]};
      *(f32x4*)(S + baseRow + 8)  = lo;
      *(f32x4*)(S + baseRow + 12) = hi;
    }
  } else {  // tail tile: predicated scalar stores, constant vector indices
#pragma unroll
    for (int m = 0; m < 8; ++m) {
      const int r0 = baseRow + m;
      const int r1 = baseRow + 8 + m;
      if (lane == 0  && r0 < nNodes) S[r0] = acc[m];
      if (lane == 16 && r1 < nNodes) S[r1] = acc[m];
    }
  }
}

__global__ __launch_bounds__(256) void score_kernel(
    const float* __restrict__ S, const int* __restrict__ src,
    const int* __restrict__ dst, float* __restrict__ out, int n) {
  const float inv = 0.08838834764831845f;  // 1/sqrt(128)
  const int base = (blockIdx.x * blockDim.x + threadIdx.x) * 4;
  if (base + 3 < n) {
    i32x4 s4 = *(const i32x4*)(src + base);
    i32x4 d4 = *(const i32x4*)(dst + base);
    f32x4 o;
    o.x = (S[s4.x] - S[d4.x]) * inv;
    o.y = (S[s4.y] - S[d4.y]) * inv;
    o.z = (S[s4.z] - S[d4.z]) * inv;
    o.w = (S[s4.w] - S[d4.w]) * inv;
    *(f32x4*)(out + base) = o;
  } else {
    for (int i = base; i < n; ++i) out[i] = (S[src[i]] - S[dst[i]]) * inv;
  }
}

extern "C" void kernel_launch(void* const* d_in, const int* in_sizes, int n_in,
                              void* d_out, int out_size, void* d_ws, size_t ws_size,
                              hipStream_t stream) {
  const float* x   = (const float*)d_in[0];  // [N, 128] f32
  const int*   src = (const int*)d_in[1];    // [T, E] int32 (JAX x64-off demotes int64)
  const int*   dst = (const int*)d_in[2];    // [T, E] int32
  float*       out = (float*)d_out;          // [T, E] f32

  const int nNodes = in_sizes[0] / IN_DIM;
  const int nTotal = out_size;               // T * E
  float* S = (float*)d_ws;                   // nNodes floats (400 KB) scratch

  const int nTiles  = (nNodes + 15) / 16;
  const int blocks1 = (nTiles + WAVES_PER_BLOCK - 1) / WAVES_PER_BLOCK;
  rowsum_wmma_kernel<<<blocks1, 32 * WAVES_PER_BLOCK, 0, stream>>>(x, S, nNodes);

  const int nQuads  = (nTotal + 3) / 4;
  const int blocks2 = (nQuads + 255) / 256;
  score_kernel<<<blocks2, 256, 0, stream>>>(S, src, dst, out, nTotal);
}